// ModuleRenderScatter_38259568672883
// MI455X (gfx1250) — compile-verified
//
#include <hip/hip_runtime.h>
#include <stdint.h>
#include <stddef.h>

// Problem shape (fixed by reference setup_inputs): image [B,3,512,512] f32,
// refocus [B,1,512,512] f32. Outputs: bokeh [B,3,512,512] then trunc(refocus)
// [B,1,512,512], concatenated flat in d_out.

#define IMG_H 512
#define IMG_W 512
#define NCH   3
#define TILE  16
#define HALO  6                      // R = int(MAX_RADIUS)+1
#define SRCW  (TILE + 2*HALO)        // 28
#define NSRC  (SRCW*SRCW)            // 784
#define NOFF  (13*13)                // 169 offsets
#define NTHR  (TILE*TILE)            // 256 threads = 8 wave32

#define K8_OVER_LN2 11.5415603271f   // 8/ln(2): exp(8x) = exp2(K*x)

typedef __attribute__((address_space(1))) int as1_int;   // global, builtin pointee type
typedef __attribute__((address_space(3))) int as3_int;   // LDS

#if defined(__has_builtin)
#if __has_builtin(__builtin_amdgcn_global_load_async_to_lds_b32)
#define HAVE_ASYNC_LDS 1
#endif
#if __has_builtin(__builtin_amdgcn_s_wait_asynccnt)
#define HAVE_ASYNC_WAIT 1
#endif
#endif
#ifndef HAVE_ASYNC_LDS
#define HAVE_ASYNC_LDS 0
#endif
#ifndef HAVE_ASYNC_WAIT
#define HAVE_ASYNC_WAIT 0
#endif

__device__ __forceinline__ void wait_async0() {
#if HAVE_ASYNC_WAIT
  __builtin_amdgcn_s_wait_asynccnt(0);
#elif defined(__gfx1250__)
  asm volatile("s_wait_asynccnt 0" ::: "memory");
#endif
}

__global__ __launch_bounds__(NTHR)
void render_scatter_kernel(const float* __restrict__ img,
                           const float* __restrict__ rfo,
                           float* __restrict__ out, int B)
{
  // LDS staging: packed {r,g,b, refocus->|r|} per source pixel (ds_load_b128),
  // precomputed {E = exp(-8r), invArea = 1/(r^2+0.2)} (ds_load_b64),
  // per-offset constants {C = exp(8*d), m-1} (uniform ds broadcast).
  __shared__ float4 sPix[NSRC];     // 12544 B
  __shared__ float2 sEIA[NSRC];     //  6272 B
  __shared__ float2 sTab[NOFF];     //  1352 B  (~20 KB total)

  const int tx  = threadIdx.x, ty = threadIdx.y;
  const int tid = ty * TILE + tx;
  const int gx0 = blockIdx.x * TILE, gy0 = blockIdx.y * TILE;
  const int bz  = blockIdx.z;

  // ---- per-offset constant table (one v_sqrt + v_exp per entry, once) ----
  if (tid < NOFF) {
    const int a = tid / 13, b = tid - a * 13;
    const float fdy = (float)(a - 6), fdx = (float)(b - 6);
    const float d  = __builtin_amdgcn_sqrtf(fdy * fdy + fdx * fdx);
    const float C  = __builtin_amdgcn_exp2f(K8_OVER_LN2 * d);     // exp(8d)
    const float m1 = fmaxf(fabsf(fdy), fabsf(fdx)) - 1.0f;        // within <=> r >= m-1
    sTab[tid] = make_float2(C, m1);
  }

  const float* __restrict__ imgB = img + (size_t)bz * ((size_t)NCH * IMG_H * IMG_W);
  const float* __restrict__ rfoB = rfo + (size_t)bz * ((size_t)IMG_H * IMG_W);
  const size_t hw = (size_t)IMG_H * IMG_W;

  // ---- stage 28x28 halo tile: async global->LDS; OOB slots zeroed ----
  for (int slot = tid; slot < NSRC; slot += NTHR) {
    const int sy = slot / SRCW;
    const int sx = slot - sy * SRCW;
    const int gy = gy0 + sy - HALO;
    const int gx = gx0 + sx - HALO;
    if ((unsigned)gy < IMG_H && (unsigned)gx < IMG_W) {
      const size_t p = (size_t)gy * IMG_W + gx;
#if HAVE_ASYNC_LDS
      float* dst = (float*)&sPix[slot];
      __builtin_amdgcn_global_load_async_to_lds_b32((as1_int*)(imgB + p),          (as3_int*)(dst + 0), 0, 0);
      __builtin_amdgcn_global_load_async_to_lds_b32((as1_int*)(imgB + p + hw),     (as3_int*)(dst + 1), 0, 0);
      __builtin_amdgcn_global_load_async_to_lds_b32((as1_int*)(imgB + p + 2 * hw), (as3_int*)(dst + 2), 0, 0);
      __builtin_amdgcn_global_load_async_to_lds_b32((as1_int*)(rfoB + p),          (as3_int*)(dst + 3), 0, 0);
#else
      sPix[slot] = make_float4(imgB[p], imgB[p + hw], imgB[p + 2 * hw], rfoB[p]);
#endif
    } else {
      // zero weight and zero color => contributes nothing (matches edge mask)
      sPix[slot] = make_float4(0.f, 0.f, 0.f, 0.f);
      sEIA[slot] = make_float2(0.f, 0.f);
    }
  }
  wait_async0();          // each wave drains its own ASYNCcnt
  __syncthreads();

  // ---- per-source-pixel precompute: r, E = exp(-8r), invArea ----
  for (int slot = tid; slot < NSRC; slot += NTHR) {
    const int sy = slot / SRCW;
    const int sx = slot - sy * SRCW;
    const int gy = gy0 + sy - HALO;
    const int gx = gx0 + sx - HALO;
    if ((unsigned)gy < IMG_H && (unsigned)gx < IMG_W) {
      float4 p4 = sPix[slot];
      const float r  = fabsf(p4.w);
      const float E  = __builtin_amdgcn_exp2f(-K8_OVER_LN2 * r);  // exp(-8r)
      const float iA = 1.0f / (r * r + 0.2f);
      p4.w = r;
      sPix[slot] = p4;
      sEIA[slot] = make_float2(E, iA);
    }
  }
  __syncthreads();

  // ---- 13x13 gather: w = [r >= m-1] * rcp(C*E + 1) * invArea ----
  float acc0 = 0.f, acc1 = 0.f, acc2 = 0.f, accW = 0.f;
#pragma unroll 1
  for (int a = 0; a < 13; ++a) {
    const int rowBase = (ty + a) * SRCW + tx;
    const int tabBase = a * 13;
#pragma unroll
    for (int b = 0; b < 13; ++b) {
      const float2 cm = sTab[tabBase + b];      // uniform -> LDS broadcast
      const float4 p  = sPix[rowBase + b];      // ds_load_b128
      const float2 ei = sEIA[rowBase + b];      // ds_load_b64
      const float sig = __builtin_amdgcn_rcpf(fmaf(cm.x, ei.x, 1.0f));
      float w = sig * ei.y;
      w = (p.w >= cm.y) ? w : 0.0f;
      acc0 = fmaf(w, p.x, acc0);
      acc1 = fmaf(w, p.y, acc1);
      acc2 = fmaf(w, p.z, acc2);
      accW += w;
    }
  }

  // ---- outputs ----
  const int gy = gy0 + ty, gx = gx0 + tx;
  const float invW = 1.0f / accW;               // accW > 0 (offset (0,0) always in)
  float* ob = out + (size_t)bz * NCH * hw + (size_t)gy * IMG_W + gx;
  ob[0]      = acc0 * invW;
  ob[hw]     = acc1 * invW;
  ob[2 * hw] = acc2 * invW;
  // refocus_dilate = trunc(refocus) (second tuple element, after bokeh block)
  out[(size_t)B * NCH * hw + (size_t)bz * hw + (size_t)gy * IMG_W + gx] =
      truncf(rfoB[(size_t)gy * IMG_W + gx]);
}

extern "C" void kernel_launch(void* const* d_in, const int* in_sizes, int n_in,
                              void* d_out, int out_size, void* d_ws, size_t ws_size,
                              hipStream_t stream) {
  (void)n_in; (void)out_size; (void)d_ws; (void)ws_size;
  const float* img = (const float*)d_in[0];
  const float* rfo = (const float*)d_in[1];
  float* out = (float*)d_out;
  const int B = in_sizes[1] / (IMG_H * IMG_W);  // refocus is [B,1,H,W]
  dim3 grid(IMG_W / TILE, IMG_H / TILE, B);
  dim3 block(TILE, TILE, 1);
  hipLaunchKernelGGL(render_scatter_kernel, grid, block, 0, stream, img, rfo, out, B);
}